// DifferentiableAAC_14663018348618
// MI455X (gfx1250) — compile-verified
//
#include <hip/hip_runtime.h>
#include <math.h>

typedef float v2f __attribute__((ext_vector_type(2)));
typedef float v8f __attribute__((ext_vector_type(8)));

namespace aac {
constexpr int KM = 1024;              // MDCT bins (M)
constexpr int KN = 2048;              // window length (N)
constexpr int KB = 8;                 // batch
constexpr int KC = 2;                 // channels
constexpr int KF = 1408;              // frames
constexpr int KT = 1440000;           // samples
constexpr int KR = KB * KC * KF;      // 22528 GEMM rows
constexpr int SA = 36;                // A-tile LDS row stride (words)
constexpr int SB = 288;               // B-tile LDS k-pair row stride (words)
constexpr float TARGET_BITS = 2730.6666667f;  // 128000*1024/48000
constexpr double PI_D = 3.14159265358979323846;
}
using namespace aac;

// -------- async global->LDS staging (GLOBAL_LOAD_ASYNC_TO_LDS, ASYNCcnt) ----
#if __has_builtin(__builtin_amdgcn_global_load_async_to_lds_b32) && \
    __has_builtin(__builtin_amdgcn_global_load_async_to_lds_b128)
#define AAC_ASYNC_LDS 1
#else
#define AAC_ASYNC_LDS 0
#endif

#if AAC_ASYNC_LDS
typedef __attribute__((address_space(1))) int as1_int;  // HIP __device__ = AS1
typedef __attribute__((address_space(3))) int as3_int;  // __shared__     = AS3
typedef int v4i_g __attribute__((vector_size(16)));
typedef __attribute__((address_space(1))) v4i_g as1_v4i;
typedef __attribute__((address_space(3))) v4i_g as3_v4i;
#endif

__device__ __forceinline__ void g2l_b32(const float* g, float* l) {
#if AAC_ASYNC_LDS
  __builtin_amdgcn_global_load_async_to_lds_b32(
      (as1_int*)(void*)g, (as3_int*)(void*)l, 0, 0);
#else
  *l = *g;
#endif
}
__device__ __forceinline__ void g2l_b128(const float* g, float* l) {
#if AAC_ASYNC_LDS
  __builtin_amdgcn_global_load_async_to_lds_b128(
      (as1_v4i*)(void*)g, (as3_v4i*)(void*)l, 0, 0);
#else
#pragma unroll
  for (int i = 0; i < 4; ++i) l[i] = g[i];
#endif
}
__device__ __forceinline__ void async_wait() {
#if AAC_ASYNC_LDS
  asm volatile("s_wait_asynccnt 0x0" ::: "memory");
#endif
}

// ---------------------------------------------------------------------------
// K0: cosine matrix (N x M), transpose (M x N), window
// ---------------------------------------------------------------------------
__global__ void aac_init_tables(float* __restrict__ Cm, float* __restrict__ CmT,
                                float* __restrict__ wtab) {
  long idx = (long)blockIdx.x * 256 + threadIdx.x;
  if (idx < (long)KN * KM) {
    int n = (int)(idx / KM);
    int k = (int)(idx % KM);
    double ph = (PI_D / (double)KM) * ((double)n + 0.5 + (double)(KM / 2)) *
                ((double)k + 0.5);
    float cv = (float)cos(ph);
    Cm[idx] = cv;
    CmT[(long)k * KN + n] = cv;
  }
  if (idx < KN) {
    wtab[idx] = sinf((float)(PI_D / (double)KN) * ((float)idx + 0.5f));
  }
}

__global__ void aac_zero_out(float* __restrict__ out, int nelem) {
  int i = blockIdx.x * 256 + threadIdx.x;
  if (i < nelem) out[i] = 0.0f;
}

// ---------------------------------------------------------------------------
// K1: forward MDCT WMMA f32 GEMM. Block(256)=8 waves -> 64x128 tile.
// ---------------------------------------------------------------------------
__global__ void aac_mdct_fwd(const float* __restrict__ audio,
                             const float* __restrict__ Cm,
                             const float* __restrict__ wtab,
                             float* __restrict__ coeffs) {
  __shared__ __align__(16) float Alds[64 * SA];
  __shared__ __align__(16) float Blds[16 * SB];

  const int tiles_n = KM / 128;  // 8
  int row0 = (blockIdx.x / tiles_n) * 64;
  int col0 = (blockIdx.x % tiles_n) * 128;
  int tid = threadIdx.x;
  int lane = tid & 31, wv = tid >> 5;
  int rt = wv >> 1, ct = wv & 1;
  int half = lane >> 4, lm = lane & 15;

  v8f acc[4];
#pragma unroll
  for (int i = 0; i < 4; ++i)
#pragma unroll
    for (int j = 0; j < 8; ++j) acc[i][j] = 0.0f;

  for (int k0 = 0; k0 < KN; k0 += 32) {
    // A chunk (64 rows x 32 K): fused framing + window (compute path)
#pragma unroll
    for (int i = 0; i < 8; ++i) {
      int e = tid + 256 * i;
      int m = e >> 5, k = e & 31;
      int r = row0 + m;
      int f = r % KF;
      int bc = r / KF;
      int gk = k0 + k;
      long t = (long)f * KM + gk - KM;
      float v = 0.0f;
      if (t >= 0 && t < KT) v = audio[(long)bc * KT + t] * wtab[gk];
      Alds[m * SA + k] = v;
    }
    // B chunk (32 K x 128 cols), k-pair interleaved, async to LDS
    if (k0 + 32 < KN)
      __builtin_prefetch(&Cm[(long)(k0 + 32) * KM + col0 + (tid & 127)], 0, 1);
#pragma unroll
    for (int i = 0; i < 16; ++i) {
      int e = tid + 256 * i;
      int k = e >> 7, n = e & 127;
      g2l_b32(&Cm[(long)(k0 + k) * KM + col0 + n],
              &Blds[(k >> 1) * SB + n * 2 + (k & 1)]);
    }
    async_wait();
    __syncthreads();

#pragma unroll
    for (int ks = 0; ks < 32; ks += 4) {
      int kk = ks + half * 2;
      v2f a = *(const v2f*)&Alds[(rt * 16 + lm) * SA + kk];
      int kp = (ks >> 1) + half;
#pragma unroll
      for (int nt = 0; nt < 4; ++nt) {
        int nl = ct * 64 + nt * 16 + lm;
        v2f b = *(const v2f*)&Blds[kp * SB + nl * 2];
        acc[nt] = __builtin_amdgcn_wmma_f32_16x16x4_f32(
            false, a, false, b, (short)0, acc[nt], false, false);
      }
    }
    __syncthreads();
  }

  int rbase = row0 + rt * 16 + half * 8;
#pragma unroll
  for (int nt = 0; nt < 4; ++nt) {
    int n = col0 + ct * 64 + nt * 16 + lm;
#pragma unroll
    for (int i = 0; i < 8; ++i) {
      coeffs[(long)(rbase + i) * KM + n] = acc[nt][i];
    }
  }
}

// ---------------------------------------------------------------------------
// K2: per-(b,f) 8-step bisection for the global gain
// ---------------------------------------------------------------------------
__global__ void aac_gain_search(const float* __restrict__ coeffs,
                                float* __restrict__ gains) {
  int bf = blockIdx.x;
  int b = bf / KF, f = bf % KF;
  int tid = threadIdx.x;

  float ax[8];
#pragma unroll
  for (int i = 0; i < 8; ++i) {
    int e = tid + 256 * i;
    int c = e >> 10, k = e & 1023;
    float x = coeffs[((long)((b * KC + c) * KF + f)) * KM + k];
    ax[i] = powf(fabsf(x), 0.75f);
  }

  __shared__ float red[256];
  __shared__ float s_lo, s_hi;
  if (tid == 0) { s_lo = 0.0f; s_hi = 120.0f; }
  __syncthreads();

  for (int it = 0; it < 8; ++it) {
    float mid = floorf((s_lo + s_hi) * 0.5f);
    float inv = exp2f(-0.1875f * mid);
    float bits = 0.0f;
#pragma unroll
    for (int i = 0; i < 8; ++i) {
      float q = rintf(ax[i] * inv);
      if (q > 0.0f) {
        int e2 = (int)((__float_as_uint(q) >> 23) & 255u) - 127;
        bits += (float)(e2 + 2);
      } else {
        bits += 1.0f;
      }
    }
    red[tid] = bits;
    __syncthreads();
    for (int s = 128; s > 0; s >>= 1) {
      if (tid < s) red[tid] += red[tid + s];
      __syncthreads();
    }
    if (tid == 0) {
      if (red[0] > TARGET_BITS) s_lo = mid + 1.0f;
      else                      s_hi = mid;
    }
    __syncthreads();
  }
  if (tid == 0) gains[bf] = s_hi;
}

// ---------------------------------------------------------------------------
// K3: quantize + dequantize in place
// ---------------------------------------------------------------------------
__global__ void aac_quantize(float* __restrict__ coeffs,
                             const float* __restrict__ gains) {
  long idx = (long)blockIdx.x * 256 + threadIdx.x;
  if (idx >= (long)KR * KM) return;
  int r = (int)(idx >> 10);
  int f = r % KF;
  int b = r / (KC * KF);
  float g = gains[b * KF + f];
  float scale = exp2f(0.25f * g);
  float x = coeffs[idx];
  float qs = powf(fabsf(x) / scale + 1e-9f, 0.75f);
  float rq = rintf(qs);
  float dq = powf(rq, 4.0f / 3.0f) * scale;
  coeffs[idx] = copysignf(dq, x);
}

// ---------------------------------------------------------------------------
// K4: inverse MDCT WMMA GEMM + fused window*2/M + overlap-add scatter
// ---------------------------------------------------------------------------
__global__ void aac_imdct_oadd(const float* __restrict__ dq,
                               const float* __restrict__ CmT,
                               const float* __restrict__ wtab,
                               float* __restrict__ out) {
  __shared__ __align__(16) float Alds[64 * SA];
  __shared__ __align__(16) float Blds[16 * SB];

  const int tiles_n = KN / 128;  // 16
  int row0 = (blockIdx.x / tiles_n) * 64;
  int col0 = (blockIdx.x % tiles_n) * 128;
  int tid = threadIdx.x;
  int lane = tid & 31, wv = tid >> 5;
  int rt = wv >> 1, ct = wv & 1;
  int half = lane >> 4, lm = lane & 15;

  v8f acc[4];
#pragma unroll
  for (int i = 0; i < 4; ++i)
#pragma unroll
    for (int j = 0; j < 8; ++j) acc[i][j] = 0.0f;

  for (int k0 = 0; k0 < KM; k0 += 32) {
    // A chunk: contiguous 32B per thread -> async b128 x2
    {
      int m = tid >> 2, k = (tid & 3) * 8;
      const float* src = dq + (long)(row0 + m) * KM + k0 + k;
      float* dst = &Alds[m * SA + k];
      g2l_b128(src, dst);
      g2l_b128(src + 4, dst + 4);
    }
    // B chunk: k-pair interleaved async b32
    if (k0 + 32 < KM)
      __builtin_prefetch(&CmT[(long)(k0 + 32) * KN + col0 + (tid & 127)], 0, 1);
#pragma unroll
    for (int i = 0; i < 16; ++i) {
      int e = tid + 256 * i;
      int k = e >> 7, n = e & 127;
      g2l_b32(&CmT[(long)(k0 + k) * KN + col0 + n],
              &Blds[(k >> 1) * SB + n * 2 + (k & 1)]);
    }
    async_wait();
    __syncthreads();

#pragma unroll
    for (int ks = 0; ks < 32; ks += 4) {
      int kk = ks + half * 2;
      v2f a = *(const v2f*)&Alds[(rt * 16 + lm) * SA + kk];
      int kp = (ks >> 1) + half;
#pragma unroll
      for (int nt = 0; nt < 4; ++nt) {
        int nl = ct * 64 + nt * 16 + lm;
        v2f b = *(const v2f*)&Blds[kp * SB + nl * 2];
        acc[nt] = __builtin_amdgcn_wmma_f32_16x16x4_f32(
            false, a, false, b, (short)0, acc[nt], false, false);
      }
    }
    __syncthreads();
  }

  const float sc = 2.0f / (float)KM;
  int rbase = row0 + rt * 16 + half * 8;
#pragma unroll
  for (int nt = 0; nt < 4; ++nt) {
    int n = col0 + ct * 64 + nt * 16 + lm;
    float wn = wtab[n] * sc;
#pragma unroll
    for (int i = 0; i < 8; ++i) {
      int r = rbase + i;
      int f = r % KF;
      int bc = r / KF;
      long t = (long)f * KM + n - KM;
      if (t >= 0 && t < KT) atomicAdd(out + (long)bc * KT + t, acc[nt][i] * wn);
    }
  }
}

// ---------------------------------------------------------------------------
extern "C" void kernel_launch(void* const* d_in, const int* in_sizes, int n_in,
                              void* d_out, int out_size, void* d_ws, size_t ws_size,
                              hipStream_t stream) {
  (void)in_sizes; (void)n_in; (void)ws_size;
  const float* audio = (const float*)d_in[0];
  float* out = (float*)d_out;

  float* ws     = (float*)d_ws;
  float* Cm     = ws;                          // N*M   (8 MB)
  float* CmT    = Cm  + (size_t)KN * KM;       // M*N   (8 MB)
  float* wtab   = CmT + (size_t)KM * KN;       // N
  float* coeffs = wtab + KN;                   // R*M   (88 MB), reused as dq
  float* gains  = coeffs + (size_t)KR * KM;    // B*F

  aac_init_tables<<<(KN * KM + 255) / 256, 256, 0, stream>>>(Cm, CmT, wtab);
  aac_zero_out<<<(out_size + 255) / 256, 256, 0, stream>>>(out, out_size);
  aac_mdct_fwd<<<(KR / 64) * (KM / 128), 256, 0, stream>>>(audio, Cm, wtab, coeffs);
  aac_gain_search<<<KB * KF, 256, 0, stream>>>(coeffs, gains);
  aac_quantize<<<(int)(((long)KR * KM + 255) / 256), 256, 0, stream>>>(coeffs, gains);
  aac_imdct_oadd<<<(KR / 64) * (KN / 128), 256, 0, stream>>>(coeffs, CmT, wtab, out);
}